// Observer_49916109914358
// MI455X (gfx1250) — compile-verified
//
#include <hip/hip_runtime.h>
#include <stdint.h>

#define ROWS 8192
#define COLS 8192
#define BRK 128                 // block rows
#define BCK 128                 // block cols
#define NRB (ROWS / BRK)        // 64
#define NCB (COLS / BCK)        // 64
#define SUB_ROWS 32             // TDM subtile rows (16 KB per subtile)
#define NSUB (BRK / SUB_ROWS)   // 4 subtiles per block
#define TPB 256                 // 8 wave32

typedef __attribute__((ext_vector_type(4))) unsigned int v4u;
typedef __attribute__((ext_vector_type(8))) int          v8i;
typedef __attribute__((ext_vector_type(4))) int          v4i;
typedef __attribute__((ext_vector_type(4))) float        v4f;

// Issue one TDM tensor load: 32x128 f32 subtile (row-major, stride = COLS)
// from global memory into LDS at byte offset lds_off.
__device__ __forceinline__ void tdm_load_subtile(const float* gbase,
                                                 uint32_t lds_off,
                                                 int row0, int col0) {
  uint64_t ga = (uint64_t)(uintptr_t)gbase +
                ((uint64_t)row0 * COLS + (uint64_t)col0) * 4ull;
  // D# group 0 (128b): count=1 | lds_addr | global_addr[56:0] | type=2
  v4u g0;
  g0.x = 1u;                                               // count=1, user mode
  g0.y = lds_off;                                          // lds_addr (bytes)
  g0.z = (uint32_t)ga;                                     // global_addr[31:0]
  g0.w = (uint32_t)((ga >> 32) & 0x1FFFFFFu) | (2u << 30); // addr[56:32], type=2
  // D# group 1 (256b)
  v8i g1;
  g1[0] = (int)(2u << 16);                                 // wg_mask=0, data_size=4B
  g1[1] = (int)(((uint32_t)COLS & 0xFFFFu) << 16);         // tensor_dim0[15:0]
  g1[2] = (int)((((uint32_t)COLS >> 16) & 0xFFFFu) |
                (((uint32_t)ROWS & 0xFFFFu) << 16));       // dim0[31:16] | dim1[15:0]
  g1[3] = (int)((((uint32_t)ROWS >> 16) & 0xFFFFu) |
                ((uint32_t)BCK << 16));                    // dim1[31:16] | tile_dim0=128
  g1[4] = SUB_ROWS;                                        // tile_dim1=32, tile_dim2=0
  g1[5] = COLS;                                            // tensor_dim0_stride[31:0]
  g1[6] = 0;                                               // stride0 hi | stride1 lo
  g1[7] = 0;
  v4i gz4 = {0, 0, 0, 0};                                  // groups 2/3 unused (2D)
  v8i gz8 = {0, 0, 0, 0, 0, 0, 0, 0};                      // extra group unused
  __builtin_amdgcn_tensor_load_to_lds(g0, g1, gz4, gz4, gz8, 0);
}

__global__ __launch_bounds__(TPB) void observer_block_qparams(
    const float* __restrict__ obs,
    float* __restrict__ scale_out,
    int* __restrict__ zp_out) {
  __shared__ float tile[2][SUB_ROWS * BCK];   // 2 x 16 KB ping-pong
  __shared__ float wmin[8], wmax[8];
  __shared__ float s_scale;
  __shared__ int   s_zp;

  const int tid  = threadIdx.x;
  const int wave = tid >> 5;
  const int lane = tid & 31;
  const int blk  = blockIdx.x;
  const int brow = blk / NCB;
  const int bcol = blk % NCB;
  const int row0 = brow * BRK;
  const int col0 = bcol * BCK;

  // LDS flat address low 32 bits == wave-relative LDS byte offset (aperture
  // base has zero low bits per ISA aperture layout).
  const uint32_t lds0 = (uint32_t)(uintptr_t)&tile[0][0];
  const uint32_t lds1 = (uint32_t)(uintptr_t)&tile[1][0];

  // Prime the pipeline: wave 0 issues subtiles 0 and 1.
  if (wave == 0) {
    tdm_load_subtile(obs, lds0, row0 + 0 * SUB_ROWS, col0);
    tdm_load_subtile(obs, lds1, row0 + 1 * SUB_ROWS, col0);
  }

  float vmin =  3.402823466e+38f;
  float vmax = -3.402823466e+38f;

#pragma unroll
  for (int s = 0; s < NSUB; ++s) {
    if (wave == 0) {
      if (s < NSUB - 1) __builtin_amdgcn_s_wait_tensorcnt(1); // subtile s landed
      else              __builtin_amdgcn_s_wait_tensorcnt(0); // last one
    }
    __syncthreads();
    // Reduce 32x128 subtile: 256 threads x 4 x float4, lane-contiguous
    // 16B accesses -> conflict-free ds_load_b128.
    const v4f* tp = (const v4f*)tile[s & 1];
#pragma unroll
    for (int k = 0; k < 4; ++k) {
      v4f v = tp[tid + TPB * k];
      vmin = fminf(vmin, fminf(fminf(v.x, v.y), fminf(v.z, v.w)));
      vmax = fmaxf(vmax, fmaxf(fmaxf(v.x, v.y), fmaxf(v.z, v.w)));
    }
    __syncthreads(); // everyone done reading buffer before TDM overwrites it
    if (wave == 0 && (s + 2) < NSUB) {
      tdm_load_subtile(obs, (s & 1) ? lds1 : lds0,
                       row0 + (s + 2) * SUB_ROWS, col0);
    }
  }

  // wave32 butterfly reduction
#pragma unroll
  for (int off = 16; off >= 1; off >>= 1) {
    vmin = fminf(vmin, __shfl_xor(vmin, off, 32));
    vmax = fmaxf(vmax, __shfl_xor(vmax, off, 32));
  }
  if (lane == 0) { wmin[wave] = vmin; wmax[wave] = vmax; }
  __syncthreads();

  if (tid == 0) {
    float bmin = wmin[0], bmax = wmax[0];
#pragma unroll
    for (int w = 1; w < 8; ++w) {
      bmin = fminf(bmin, wmin[w]);
      bmax = fmaxf(bmax, wmax[w]);
    }
    float rng = bmax - bmin;
    float sc; int zp;
    if (rng == 0.0f) {
      sc = 1.0f; zp = 0;
    } else {
      sc = rng * (1.0f / 255.0f);                  // rng / (qmax - qmin)
      zp = (int)rintf(-128.0f - bmin / sc);        // round-to-nearest-even
    }
    s_scale = sc;
    s_zp = zp;
  }
  __syncthreads();

  const float sc = s_scale;
  const int   z  = s_zp;
  v4f sv = {sc, sc, sc, sc};
  v4i zv = {z, z, z, z};

  // Replicate over the 128x128 region: each wave writes one full row per pass
  // (32 lanes x 16B = 512B contiguous), non-temporal (streamed, never reread).
#pragma unroll
  for (int pass = 0; pass < BRK / 8; ++pass) {
    const int row = pass * 8 + wave;
    const size_t base = (size_t)(row0 + row) * COLS + (size_t)col0 + lane * 4;
    __builtin_nontemporal_store(sv, (v4f*)(scale_out + base));
    __builtin_nontemporal_store(zv, (v4i*)(zp_out + base));
  }
}

extern "C" void kernel_launch(void* const* d_in, const int* in_sizes, int n_in,
                              void* d_out, int out_size, void* d_ws, size_t ws_size,
                              hipStream_t stream) {
  (void)in_sizes; (void)n_in; (void)d_ws; (void)ws_size; (void)out_size;
  const float* obs = (const float*)d_in[0];
  float* scale_out = (float*)d_out;
  int*   zp_out    = (int*)d_out + (size_t)ROWS * (size_t)COLS; // second output
  observer_block_qparams<<<dim3(NRB * NCB), dim3(TPB), 0, stream>>>(
      obs, scale_out, zp_out);
}